// BiologicalMemory_2602750181563
// MI455X (gfx1250) — compile-verified
//
#include <hip/hip_runtime.h>
#include <hip/hip_bf16.h>
#include <math.h>

typedef float v2f __attribute__((ext_vector_type(2)));
typedef float v8f __attribute__((ext_vector_type(8)));

#define DIM   2048
#define NMEM  60000
#define EPSF  1e-8f
#define WPB   8      // waves per block (256 threads, wave32)
#define RPW   16     // rows per wave (one 16x16 WMMA C tile)

__device__ __forceinline__ v8f wmma_acc(v2f a, v2f b, v8f c) {
  // D = A(16x4 f32) * B(4x16 f32) + C ; emits v_wmma_f32_16x16x4_f32
  return __builtin_amdgcn_wmma_f32_16x16x4_f32(false, a, false, b, (short)0, c,
                                               false, false);
}

// y[r] = dot(W[r,:], x) + bias[r] for 16 rows per wave via WMMA.
// A layout (f32 16x4): lane = row (lane&15); VGPR pair = K {2h, 2h+1}, h=lane>>4.
// B broadcast: every N column holds the same x slice -> every D column = GEMV.
__global__ void __launch_bounds__(256)
gemv_wmma(const float* __restrict__ W, const float* __restrict__ x,
          const float* __restrict__ bias, float* __restrict__ y, int nrows) {
  const int tid  = threadIdx.x;
  const int wave = tid >> 5;
  const int lane = tid & 31;
  const int r0   = (blockIdx.x * WPB + wave) * RPW;
  if (r0 >= nrows) return;                 // wave-uniform: EXEC stays all-ones
  const int half = lane >> 4;
  const int j    = lane & 15;

  const float* arow = W + (size_t)(r0 + j) * DIM + 2 * half;
  const float* bvec = x + 2 * half;

  v8f c = {0.f, 0.f, 0.f, 0.f, 0.f, 0.f, 0.f, 0.f};
#pragma unroll 4
  for (int k = 0; k < DIM; k += 4) {
    v2f a = *(const v2f*)(arow + k);
    v2f b = *(const v2f*)(bvec + k);
    c = wmma_acc(a, b, c);
  }
  if (j < 8) {                             // lanes {0..7, 16..23} write rows 0..15
    const int rl = half * 8 + j;           // C tile: vgpr v, half h -> row v + 8h
    float dot = 0.f;
#pragma unroll
    for (int v = 0; v < 8; ++v)
      if (j == v) dot = c[v];
    y[r0 + rl] = dot + bias[r0 + rl];
  }
}

// qn = q / max(||q||, eps), in place, single block of 256 (8 floats/thread).
__global__ void __launch_bounds__(256) normalize_q(float* __restrict__ q) {
  __shared__ float red[256];
  __shared__ float invn;
  const int tid = threadIdx.x;
  float4* q4 = (float4*)q;
  float4 a = q4[2 * tid], b = q4[2 * tid + 1];
  red[tid] = a.x * a.x + a.y * a.y + a.z * a.z + a.w * a.w +
             b.x * b.x + b.y * b.y + b.z * b.z + b.w * b.w;
  __syncthreads();
  for (int s = 128; s > 0; s >>= 1) {
    if (tid < s) red[tid] += red[tid + s];
    __syncthreads();
  }
  if (tid == 0) invn = 1.f / fmaxf(sqrtf(red[0]), EPSF);
  __syncthreads();
  const float iv = invn;
  a.x *= iv; a.y *= iv; a.z *= iv; a.w *= iv;
  b.x *= iv; b.y *= iv; b.z *= iv; b.w *= iv;
  q4[2 * tid] = a;
  q4[2 * tid + 1] = b;
}

// Dominant pass: one streaming read of `memories` computes dot(m, qn) on the
// matrix pipe (WMMA) and ||m||^2 on the VALU (co-executing fmas), then the
// weighted-similarity block argmax.
__global__ void __launch_bounds__(256)
sim_argmax(const float* __restrict__ mem, const float* __restrict__ qn,
           const float* __restrict__ imp, float* __restrict__ bval,
           int* __restrict__ bidx) {
  __shared__ float sv[256];
  __shared__ int   si[256];
  const int tid  = threadIdx.x;
  const int wave = tid >> 5;
  const int lane = tid & 31;
  const int r0   = (blockIdx.x * WPB + wave) * RPW;

  float wbest = -__builtin_inff();
  int   gbest = 0x7fffffff;

  if (r0 < NMEM) {                         // wave-uniform (NMEM % 16 == 0)
    const int half = lane >> 4;
    const int j    = lane & 15;
    const float* arow = mem + (size_t)(r0 + j) * DIM + 2 * half;
    const float* bvec = qn + 2 * half;

    v8f   c  = {0.f, 0.f, 0.f, 0.f, 0.f, 0.f, 0.f, 0.f};
    float sq = 0.f;                        // this lane's half of row j's ||m||^2
#pragma unroll 4
    for (int k = 0; k < DIM; k += 4) {
      v2f a = *(const v2f*)(arow + k);
      v2f b = *(const v2f*)(bvec + k);
      sq = fmaf(a.x, a.x, sq);
      sq = fmaf(a.y, a.y, sq);
      c = wmma_acc(a, b, c);
    }
    // row r norm = partial(lane r) + partial(lane r+16)
    const int   rl = half * 8 + (j & 7);
    const float s0 = __shfl(sq, rl, 32);
    const float s1 = __shfl(sq, rl + 16, 32);
    if (j < 8) {
      float dot = 0.f;
#pragma unroll
      for (int v = 0; v < 8; ++v)
        if (j == v) dot = c[v];
      const int   g   = r0 + rl;
      const float sim = dot / fmaxf(sqrtf(s0 + s1), EPSF);
      wbest = sim * imp[g];
      gbest = g;
    }
  }
  sv[tid] = wbest;
  si[tid] = gbest;
  __syncthreads();
  for (int s = 128; s > 0; s >>= 1) {
    if (tid < s) {
      const float ov = sv[tid + s];
      const int   oi = si[tid + s];
      if (ov > sv[tid] || (ov == sv[tid] && oi < si[tid])) { sv[tid] = ov; si[tid] = oi; }
    }
    __syncthreads();
  }
  if (tid == 0) { bval[blockIdx.x] = sv[0]; bidx[blockIdx.x] = si[0]; }
}

// Reduce per-block winners, then gather memories[best] into ws.
__global__ void __launch_bounds__(256)
final_pick(const float* __restrict__ bval, const int* __restrict__ bidx,
           int nblocks, const float* __restrict__ mem, float* __restrict__ emb) {
  __shared__ float sv[256];
  __shared__ int   si[256];
  __shared__ int   best;
  const int tid = threadIdx.x;
  float v  = -__builtin_inff();
  int   id = 0x7fffffff;
  for (int i = tid; i < nblocks; i += 256) {
    const float bv = bval[i];
    const int   bi = bidx[i];
    if (bv > v || (bv == v && bi < id)) { v = bv; id = bi; }
  }
  sv[tid] = v;
  si[tid] = id;
  __syncthreads();
  for (int s = 128; s > 0; s >>= 1) {
    if (tid < s) {
      const float ov = sv[tid + s];
      const int   oi = si[tid + s];
      if (ov > sv[tid] || (ov == sv[tid] && oi < si[tid])) { sv[tid] = ov; si[tid] = oi; }
    }
    __syncthreads();
  }
  if (tid == 0) best = si[0];
  __syncthreads();
  const float* src = mem + (size_t)best * DIM;
  for (int i = tid; i < DIM; i += 256) emb[i] = src[i];
}

extern "C" void kernel_launch(void* const* d_in, const int* in_sizes, int n_in,
                              void* d_out, int out_size, void* d_ws, size_t ws_size,
                              hipStream_t stream) {
  const float* query      = (const float*)d_in[0];
  const float* memories   = (const float*)d_in[1];
  const float* importance = (const float*)d_in[2];
  const float* W_enc      = (const float*)d_in[3];
  const float* b_enc      = (const float*)d_in[4];
  const float* W_dec      = (const float*)d_in[5];
  const float* b_dec      = (const float*)d_in[6];
  float* out = (float*)d_out;

  // workspace layout (floats): qn[2048] | emb[2048] | bval[512] | bidx[512]
  float* wsF  = (float*)d_ws;
  float* qn   = wsF;
  float* emb  = wsF + DIM;
  float* bval = wsF + 2 * DIM;
  int*   bidx = (int*)(wsF + 2 * DIM + 512);

  constexpr int SIM_BLOCKS  = ((NMEM / RPW) + WPB - 1) / WPB;   // 469
  constexpr int GEMV_BLOCKS = ((DIM / RPW) + WPB - 1) / WPB;    // 16

  gemv_wmma<<<GEMV_BLOCKS, 256, 0, stream>>>(W_enc, query, b_enc, qn, DIM);
  normalize_q<<<1, 256, 0, stream>>>(qn);
  sim_argmax<<<SIM_BLOCKS, 256, 0, stream>>>(memories, qn, importance, bval, bidx);
  final_pick<<<1, 256, 0, stream>>>(bval, bidx, SIM_BLOCKS, memories, emb);
  gemv_wmma<<<GEMV_BLOCKS, 256, 0, stream>>>(W_dec, emb, b_dec, out, DIM);
}